// SAGE_Body_26671746908237
// MI455X (gfx1250) — compile-verified
//
#include <hip/hip_runtime.h>

typedef __attribute__((ext_vector_type(2))) float v2f;
typedef __attribute__((ext_vector_type(8))) float v8f;

#define N_NODES   100000
#define NFEAT     256
#define NHID      256
#define KTOT      512          // fused K: 256 (x @ W_self) + 256 (h_neigh @ W_neigh)
#define ROWS_BLK  32           // rows (nodes) per block; 100000 % 32 == 0 -> 3125 blocks

// ---------------------------------------------------------------------------
// Kernel 0: zero agg (aliased onto d_out) and deg
// ---------------------------------------------------------------------------
__global__ void sage_zero_kernel(float* __restrict__ agg, float* __restrict__ deg) {
    size_t i = (size_t)blockIdx.x * blockDim.x + threadIdx.x;
    size_t n4 = (size_t)N_NODES * NHID / 4;          // 6,400,000 float4s
    if (i < n4) {
        float4 z; z.x = 0.f; z.y = 0.f; z.z = 0.f; z.w = 0.f;
        ((float4*)agg)[i] = z;
    }
    if (i < (size_t)N_NODES) deg[i] = 0.f;
}

// ---------------------------------------------------------------------------
// Kernel 1: edge scatter-add. One wave32 per edge; each lane handles 8 floats.
// ---------------------------------------------------------------------------
__global__ void sage_scatter_kernel(const float* __restrict__ x,
                                    const int* __restrict__ esrc,
                                    const int* __restrict__ edst,
                                    float* __restrict__ agg,
                                    float* __restrict__ deg,
                                    int n_edges) {
    int gid  = blockIdx.x * blockDim.x + threadIdx.x;
    int edge = gid >> 5;
    int lane = gid & 31;
    if (edge >= n_edges) return;

    int s = esrc[edge];
    int d = edst[edge];

    const float4* xs = (const float4*)(x + (size_t)s * NFEAT) + lane * 2;
    float4 a = xs[0];
    float4 b = xs[1];

    float* ap = agg + (size_t)d * NFEAT + lane * 8;
    atomicAdd(ap + 0, a.x); atomicAdd(ap + 1, a.y);
    atomicAdd(ap + 2, a.z); atomicAdd(ap + 3, a.w);
    atomicAdd(ap + 4, b.x); atomicAdd(ap + 5, b.y);
    atomicAdd(ap + 6, b.z); atomicAdd(ap + 7, b.w);

    if (lane == 0) atomicAdd(deg + d, 1.0f);
}

// ---------------------------------------------------------------------------
// Kernel 2: fused GEMM  out = relu( [x | agg/deg] @ [W_self ; W_neigh] + b )
// Block = 256 threads = 8 waves, covers 32 rows x 256 cols.
// Wave w: rowTile = w>>2 (16 rows), colBase = (w&3)*64 (4 x 16x16 WMMA tiles).
// A tile staged K-major in LDS: lA[k*32 + r], exactly 64 KB.
// K-loop split at 256 so each half is a branch-free pointer-increment loop.
// ---------------------------------------------------------------------------
__global__ __launch_bounds__(256)
void sage_gemm_kernel(const float* __restrict__ x,
                      const float* __restrict__ Wself,
                      const float* __restrict__ Wneigh,
                      const float* __restrict__ bias,
                      float* __restrict__ out,          // aliases agg (read-then-write, row-exclusive)
                      const float* __restrict__ deg) {
    __shared__ float lA[KTOT * ROWS_BLK];               // 512*32*4B = 65536 B

    const int tid  = threadIdx.x;
    const int r0   = blockIdx.x * ROWS_BLK;

    // ---- stage A: x cols [0,256) and scaled agg cols [256,512), K-major ----
    // thread t owns row r = t & 31 (constant across iters) -> load deg once.
    const int r    = tid & 31;
    const float dg = deg[r0 + r];
    const float inv = 1.0f / fmaxf(dg, 1.0f);

#pragma unroll
    for (int it = 0; it < 16; ++it) {
        int idx = tid + it * 256;        // 0..4095
        int c4  = idx >> 5;              // 0..127 (float4 index along K)
        int c   = c4 * 4;
        float4 v;
        if (c < NFEAT) {
            v = *(const float4*)(x + (size_t)(r0 + r) * NFEAT + c);
        } else {
            v = *(const float4*)(out + (size_t)(r0 + r) * NHID + (c - NFEAT));
            v.x *= inv; v.y *= inv; v.z *= inv; v.w *= inv;
        }
        lA[(c + 0) * ROWS_BLK + r] = v.x;
        lA[(c + 1) * ROWS_BLK + r] = v.y;
        lA[(c + 2) * ROWS_BLK + r] = v.z;
        lA[(c + 3) * ROWS_BLK + r] = v.w;
    }
    __syncthreads();

    // ---- per-wave WMMA main loop ----
    const int wid     = tid >> 5;
    const int lane    = tid & 31;
    const int half    = lane >> 4;       // 0: K+{0,1}, 1: K+{2,3}  (ISA 16x4 f32 A layout)
    const int m       = lane & 15;
    const int rowTile = wid >> 2;        // 0..1
    const int colBase = (wid & 3) * 64;  // 0,64,128,192

    const int c0 = colBase + m;          // tile t column = c0 + 16*t
    const int mr = rowTile * 16 + m;     // this lane's A row within block

    v8f acc0 = {}, acc1 = {}, acc2 = {}, acc3 = {};

#pragma unroll
    for (int hsel = 0; hsel < 2; ++hsel) {
        // weight base with lane K-offset (2*half rows) folded in
        const float* Wl = (hsel == 0 ? Wself : Wneigh) + (size_t)(2 * half) * NHID + c0;
        // LDS base for this half's K range, with lane offsets folded in
        const float* la = &lA[((hsel * NFEAT) + 2 * half) * ROWS_BLK + mr];

#pragma unroll 2
        for (int k = 0; k < NFEAT; k += 4) {
            v2f a;
            a.x = la[(k + 0) * ROWS_BLK];
            a.y = la[(k + 1) * ROWS_BLK];
            const float* Wk = Wl + (size_t)k * NHID;

            v2f b0; b0.x = Wk[0];        b0.y = Wk[NHID];
            v2f b1; b1.x = Wk[16];       b1.y = Wk[NHID + 16];
            v2f b2; b2.x = Wk[32];       b2.y = Wk[NHID + 32];
            v2f b3; b3.x = Wk[48];       b3.y = Wk[NHID + 48];

            acc0 = __builtin_amdgcn_wmma_f32_16x16x4_f32(false, a, false, b0, (short)0, acc0, false, false);
            acc1 = __builtin_amdgcn_wmma_f32_16x16x4_f32(false, a, false, b1, (short)0, acc1, false, false);
            acc2 = __builtin_amdgcn_wmma_f32_16x16x4_f32(false, a, false, b2, (short)0, acc2, false, false);
            acc3 = __builtin_amdgcn_wmma_f32_16x16x4_f32(false, a, false, b3, (short)0, acc3, false, false);
        }
    }

    // ---- epilogue: +bias, ReLU, store (C layout: VGPR j -> M=j / j+8) ----
#pragma unroll
    for (int t = 0; t < 4; ++t) {
        const v8f acc = (t == 0) ? acc0 : (t == 1) ? acc1 : (t == 2) ? acc2 : acc3;
        const int col = c0 + t * 16;
        const float bv = bias[col];
#pragma unroll
        for (int j = 0; j < 8; ++j) {
            int row = r0 + rowTile * 16 + j + 8 * half;
            float v = acc[j] + bv;
            out[(size_t)row * NHID + col] = fmaxf(v, 0.0f);
        }
    }
}

// ---------------------------------------------------------------------------
extern "C" void kernel_launch(void* const* d_in, const int* in_sizes, int n_in,
                              void* d_out, int out_size, void* d_ws, size_t ws_size,
                              hipStream_t stream) {
    const float* x     = (const float*)d_in[0];
    const float* Wself = (const float*)d_in[1];
    const float* Wneigh= (const float*)d_in[2];
    const float* bias  = (const float*)d_in[3];
    const int*   esrc  = (const int*)d_in[4];
    const int*   edst  = (const int*)d_in[5];
    float*       out   = (float*)d_out;          // also agg scratch (in-place)
    float*       deg   = (float*)d_ws;           // 100000 floats = 400 KB
    const int n_edges  = in_sizes[4];

    {   // zero agg (= d_out) and deg
        size_t n4 = (size_t)N_NODES * NHID / 4;
        int blocks = (int)((n4 + 255) / 256);
        sage_zero_kernel<<<blocks, 256, 0, stream>>>(out, deg);
    }
    {   // scatter-add: one wave per edge
        long long threads = (long long)n_edges * 32;
        int blocks = (int)((threads + 255) / 256);
        sage_scatter_kernel<<<blocks, 256, 0, stream>>>(x, esrc, edst, out, deg, n_edges);
    }
    {   // fused dual-GEMM + bias + relu
        sage_gemm_kernel<<<N_NODES / ROWS_BLK, 256, 0, stream>>>(
            x, Wself, Wneigh, bias, out, deg);
    }
}